// ButterflyConv2d_34806414967447
// MI455X (gfx1250) — compile-verified
//
#include <hip/hip_runtime.h>
#include <stdint.h>

// Butterfly grouped-conv (4 fused layers) for MI455X / gfx1250, wave32.
// x: [32, 256, 56, 56] f32 ; w_l: [64, 4, 4] f32 ; y: [32, 256, 56, 56] f32.
// Fully fused: read x once, write y once (~206 MB => ~8.8us @ 23.3 TB/s roofline).
// Math on V_WMMA_F32_16X16X4_F32 (exact f32); fills use the CDNA5 async
// memory->LDS path (GLOBAL_LOAD_ASYNC_TO_LDS_B128) with double-buffered
// per-tile prefetch. All LDS/global addressing is immediate-offset from a few
// per-lane bases (butterfly perms are affine in the lane-half bit), keeping
// VGPR pressure low.

typedef float v2f __attribute__((ext_vector_type(2)));
typedef float v8f __attribute__((ext_vector_type(8)));
typedef int   v4i __attribute__((vector_size(4 * sizeof(int))));
typedef __attribute__((address_space(1))) v4i* gv4i_p;   // global
typedef __attribute__((address_space(3))) v4i* lv4i_p;   // LDS

#define SPATIAL 3136          // 56*56
#define ROWSTR  20            // LDS row stride (floats): 16 cols + 4 pad, 80B (16B aligned)
#define NT      4             // tiles per wave (196 % 4 == 0 -> run stays in one batch)
#define TPB     196           // 16-wide spatial tiles per batch image

// ---------- CDNA5 async memory->LDS helpers ----------
__device__ __forceinline__ void async_b128(const float* g, float* l) {
#if __has_builtin(__builtin_amdgcn_global_load_async_to_lds_b128)
  __builtin_amdgcn_global_load_async_to_lds_b128(
      (gv4i_p)(uintptr_t)g, (lv4i_p)(uintptr_t)l, /*offset=*/0, /*cpol=*/0);
#else
  const uint32_t lo = (uint32_t)(uintptr_t)l;
  asm volatile("global_load_async_to_lds_b128 %0, %1, off"
               :: "v"(lo), "v"((uint64_t)(uintptr_t)g) : "memory");
#endif
}

template <int N>
__device__ __forceinline__ void wait_async() {
#if __has_builtin(__builtin_amdgcn_s_wait_asynccnt)
  __builtin_amdgcn_s_wait_asynccnt((unsigned short)N);
#else
  asm volatile("s_wait_asynccnt %0" :: "i"(N) : "memory");
#endif
}

// LDS row (h=0 half) that layer l reads for logical input channel c (c % 4 == 0).
// Layer 0 reads identity; layer l>=1 reads PERMS[l-1][c]. For c % 4 == 0 the
// perms satisfy perm(c+1) = perm(c)+4 and perm(c+2) = perm(c)+8, which is
// folded into per-lane base pointers below.
__device__ __forceinline__ constexpr int perm_row0(int l, int c) {
  switch (l) {
    case 1: return (c & ~15) | ((c >> 2) & 3);              // (c&3)==0
    case 2: return (c & ~63) | ((c & 15) << 2) | ((c >> 4) & 3);
    case 3: return            ((c & 63) << 2) | ((c >> 6) & 3);
    default: return c;
  }
}

// Issue 32 async b128 loads: one 256ch x 16col activation tile -> LDS buffer.
__device__ __forceinline__ void prefetch_tile(const float* xb, int sb, float* buf, int lane) {
  #pragma unroll
  for (int c8 = 0; c8 < 8; ++c8) {
    const int c = c8 * 32 + lane;                 // this lane owns channel row c
    const float* g = xb + (size_t)c * SPATIAL + sb;
    float* l = buf + c * ROWSTR;
    async_b128(g + 0,  l + 0);
    async_b128(g + 4,  l + 4);
    async_b128(g + 8,  l + 8);
    async_b128(g + 12, l + 12);
  }
}

__global__ __launch_bounds__(32) void butterfly_fused_kernel(
    const float* __restrict__ x,
    const float* __restrict__ w0, const float* __restrict__ w1,
    const float* __restrict__ w2, const float* __restrict__ w3,
    float* __restrict__ y)
{
  __shared__ __align__(16) float wlds[4][1024];           // 16 KB: all 4 layers' weights
  __shared__ __align__(16) float act[2][256 * ROWSTR];    // 2 x 20 KB: ping-pong tiles

  const int lane = threadIdx.x;   // single wave32 per block
  const int col  = lane & 15;     // spatial column / WMMA N index
  const int h    = lane >> 4;     // lane half

  // ---- weights -> LDS (single wave: DS in-order, no barrier needed) ----
  {
    const float* wsrc[4] = {w0, w1, w2, w3};
    for (int l = 0; l < 4; ++l) {
      const float4* s = (const float4*)wsrc[l];
      float4* d = (float4*)wlds[l];
      for (int i = lane; i < 256; i += 32) d[i] = s[i];
    }
  }

  // ---- tile run: NT consecutive 16-column tiles of one batch image ----
  const int t0  = blockIdx.x * NT;
  const int b   = t0 / TPB;
  const int sb0 = (t0 % TPB) * 16;
  const float* xb = x + (size_t)b * 256 * SPATIAL;
  float*       yb = y + (size_t)b * 256 * SPATIAL;

  prefetch_tile(xb, sb0, act[0], lane);

  #pragma unroll 1
  for (int i = 0; i < NT; ++i) {
    const int sb = sb0 + i * 16;
    float* aw = act[i & 1];

    if (i + 1 < NT) {
      prefetch_tile(xb, sb + 16, act[(i + 1) & 1], lane);  // overlap with compute
      wait_async<32>();   // async loads complete in order: tile i is now resident
    } else {
      wait_async<0>();
    }

    v8f acc[16];  // 16 chunks x (16ch x 16sp) f32 accumulators

    #pragma unroll
    for (int l = 0; l < 4; ++l) {
      // Per-lane base pointers; every access below is a compile-time offset.
      const int dh = (l == 0) ? 2 : 8;   // row step between lane halves
      const int dy = (l == 0) ? 1 : 4;   // row step for second K element
      const float* ab = aw + (h * dh) * ROWSTR + col;                       // B gathers
      const float* wb = &wlds[l][((col >> 2) << 4) | ((col & 3) << 2) | (2 * h)];

      // Phase 1: compute all 16 output chunks of this layer (reads only).
      #pragma unroll
      for (int m = 0; m < 16; ++m) {
        // Lane's A-row r=col is nonzero only in step kk==col>>2, holding
        // w_l[group = 4m + (col>>2)][o = col&3][k = 2h + j]  (one b64, imm offset).
        const v2f wv = *(const v2f*)&wb[m << 6];

        v8f d = {};
        #pragma unroll
        for (int kk = 0; kk < 4; ++kk) {
          v2f a;
          a.x = ((col >> 2) == kk) ? wv.x : 0.0f;
          a.y = ((col >> 2) == kk) ? wv.y : 0.0f;
          const int r0 = perm_row0(l, 16 * m + 4 * kk);   // constant after unroll
          v2f bv;
          bv.x = ab[r0 * ROWSTR];
          bv.y = ab[(r0 + dy) * ROWSTR];
          d = __builtin_amdgcn_wmma_f32_16x16x4_f32(
              /*neg_a=*/false, a, /*neg_b=*/false, bv,
              /*c_mod=*/(short)0, d, /*reuse_a=*/false, /*reuse_b=*/false);
        }
        acc[m] = d;
      }

      // Phase 2: write back (DS ops are in-order per wave -> no barrier needed).
      if (l < 3) {
        float* ob = aw + (8 * h) * ROWSTR + col;
        #pragma unroll
        for (int m = 0; m < 16; ++m)
          #pragma unroll
          for (int dd = 0; dd < 8; ++dd)
            ob[(16 * m + dd) * ROWSTR] = acc[m][dd];
      } else {
        float* gb = yb + (size_t)(8 * h) * SPATIAL + sb + col;
        #pragma unroll
        for (int m = 0; m < 16; ++m)
          #pragma unroll
          for (int dd = 0; dd < 8; ++dd)
            gb[(size_t)(16 * m + dd) * SPATIAL] = acc[m][dd];
      }
    }
  }
}

extern "C" void kernel_launch(void* const* d_in, const int* in_sizes, int n_in,
                              void* d_out, int out_size, void* d_ws, size_t ws_size,
                              hipStream_t stream) {
  (void)in_sizes; (void)n_in; (void)out_size; (void)d_ws; (void)ws_size;
  const float* x  = (const float*)d_in[0];
  const float* w0 = (const float*)d_in[1];
  const float* w1 = (const float*)d_in[2];
  const float* w2 = (const float*)d_in[3];
  const float* w3 = (const float*)d_in[4];
  float* y = (float*)d_out;

  // 32 batches * 196 tiles = 6272 wave-tiles; one wave32 per block, NT tiles each.
  butterfly_fused_kernel<<<6272 / NT, 32, 0, stream>>>(x, w0, w1, w2, w3, y);
}